// ColbertLoss_82575041233218
// MI455X (gfx1250) — compile-verified
//
#include <hip/hip_runtime.h>

typedef __attribute__((ext_vector_type(2))) float v2f;
typedef __attribute__((ext_vector_type(8))) float v8f;

#define NB 128   // batch (and in-batch negatives)
#define NS 32    // query tokens
#define ND 128   // doc tokens
#define NH 128   // hidden dim
#define INV_TEMP 50.0f   // 1 / 0.02
#define WAVES_PER_BLOCK 16

__device__ __forceinline__ v8f wmma_f32_4(v2f a, v2f b, v8f c) {
  // D(16x16,f32) = A(16x4,f32) * B(4x16,f32) + C
  return __builtin_amdgcn_wmma_f32_16x16x4_f32(
      /*neg_a=*/false, a, /*neg_b=*/false, b,
      /*c_mod=*/(short)0, c, /*reuse_a=*/false, /*reuse_b=*/false);
}

// One wave per (b,c) pair: score[b][c] = sum_s max_d (q[b,s,:] . p[c,d,:]).
// All 16 waves of a block share the same c, so the dominant B-operand (p[c],
// 64 KB) is fetched once into the WGP$ and broadcast to every wave.
__global__ __launch_bounds__(WAVES_PER_BLOCK * 32) void colbert_scores_kernel(
    const float* __restrict__ q, const float* __restrict__ p,
    float* __restrict__ scores) {
  const int lane = threadIdx.x & 31;
  const int wave = threadIdx.x >> 5;
  const int pair = blockIdx.x * WAVES_PER_BLOCK + wave;  // 1024 blocks * 16 waves
  const int c = pair >> 7;          // constant within a block (128/16 = 8 blocks per c)
  const int b = pair & (NB - 1);    // differs per wave

  const float* __restrict__ qb = q + (size_t)b * NS * NH;
  const float* __restrict__ pc = p + (size_t)c * ND * NH;

  // WMMA f32 16x16x4 A/B fragment addressing:
  // lane holds row (M or N) = lane%16, K = 2*(lane/16) + {0,1}  -> one float2 load
  const int row = lane & 15;
  const int kh  = (lane >> 4) << 1;

  v8f rmax0, rmax1;  // per-lane row-max for M-tile 0 (rows 0-15) and 1 (rows 16-31)
#pragma unroll
  for (int r = 0; r < 8; ++r) { rmax0[r] = -3.0e38f; rmax1[r] = -3.0e38f; }

  const float* qa0 = qb + (size_t)row * NH + kh;          // M-tile 0
  const float* qa1 = qb + (size_t)(16 + row) * NH + kh;   // M-tile 1

#pragma unroll
  for (int ng = 0; ng < 2; ++ng) {          // two groups of 4 N-tiles (64 doc tokens each)
    const float* pb = pc + (size_t)(ng * 64 + row) * NH + kh;
    v8f acc0[4], acc1[4];
#pragma unroll
    for (int n = 0; n < 4; ++n)
#pragma unroll
      for (int r = 0; r < 8; ++r) { acc0[n][r] = 0.f; acc1[n][r] = 0.f; }

    for (int k0 = 0; k0 < NH; k0 += 4) {    // K loop: 32 WMMA steps
      v2f a0 = *(const v2f*)(qa0 + k0);
      v2f a1 = *(const v2f*)(qa1 + k0);
#pragma unroll
      for (int n = 0; n < 4; ++n) {
        v2f bf = *(const v2f*)(pb + (size_t)(n * 16) * NH + k0);
        acc0[n] = wmma_f32_4(a0, bf, acc0[n]);
        acc1[n] = wmma_f32_4(a1, bf, acc1[n]);
      }
    }
    // fold this group's 4 N-tiles into the running row-max (elementwise, same lane)
#pragma unroll
    for (int n = 0; n < 4; ++n)
#pragma unroll
      for (int r = 0; r < 8; ++r) {
        rmax0[r] = fmaxf(rmax0[r], acc0[n][r]);
        rmax1[r] = fmaxf(rmax1[r], acc1[n][r]);
      }
  }

  // C layout: element (M = r + 8*(lane/16), N = lane%16). Row max -> reduce across
  // the 16 lanes of each half-wave (xor masks 1..8 stay within a half).
#pragma unroll
  for (int mask = 1; mask <= 8; mask <<= 1) {
#pragma unroll
    for (int r = 0; r < 8; ++r) {
      rmax0[r] = fmaxf(rmax0[r], __shfl_xor(rmax0[r], mask, 32));
      rmax1[r] = fmaxf(rmax1[r], __shfl_xor(rmax1[r], mask, 32));
    }
  }
  // half 0 now holds rows {0-7, 16-23}, half 1 rows {8-15, 24-31}; sum and merge halves
  float partial = 0.f;
#pragma unroll
  for (int r = 0; r < 8; ++r) partial += rmax0[r] + rmax1[r];
  partial += __shfl_xor(partial, 16, 32);

  if (lane == 0) scores[b * NB + c] = partial;
}

// One wave: cross-entropy over the 128x128 score matrix (diagonal = positives)
__global__ void colbert_loss_kernel(const float* __restrict__ scores,
                                    float* __restrict__ out) {
  const int lane = threadIdx.x;  // 32 threads
  float acc = 0.f;
  for (int b = 0; b < NB; ++b) {
    const float* row = scores + b * NB;
    float v0 = row[lane]      * INV_TEMP;
    float v1 = row[lane + 32] * INV_TEMP;
    float v2 = row[lane + 64] * INV_TEMP;
    float v3 = row[lane + 96] * INV_TEMP;
    float m = fmaxf(fmaxf(v0, v1), fmaxf(v2, v3));
#pragma unroll
    for (int mask = 16; mask >= 1; mask >>= 1)
      m = fmaxf(m, __shfl_xor(m, mask, 32));
    float s = __expf(v0 - m) + __expf(v1 - m) + __expf(v2 - m) + __expf(v3 - m);
#pragma unroll
    for (int mask = 16; mask >= 1; mask >>= 1)
      s += __shfl_xor(s, mask, 32);
    float diag = row[b] * INV_TEMP;
    acc += -(diag - m - __logf(s));   // -log_softmax at the diagonal
  }
  if (lane == 0) out[0] = acc / (float)NB;
}

extern "C" void kernel_launch(void* const* d_in, const int* in_sizes, int n_in,
                              void* d_out, int out_size, void* d_ws, size_t ws_size,
                              hipStream_t stream) {
  const float* q = (const float*)d_in[0];   // [128, 32, 128] f32
  const float* p = (const float*)d_in[1];   // [128, 128, 128] f32
  float* out = (float*)d_out;               // scalar loss
  float* scores = (float*)d_ws;             // 128*128 f32 = 64 KB scratch

  colbert_scores_kernel<<<(NB * NB) / WAVES_PER_BLOCK, WAVES_PER_BLOCK * 32, 0,
                          stream>>>(q, p, scores);
  colbert_loss_kernel<<<1, 32, 0, stream>>>(scores, out);
}